// PointerNetwork_1760936591897
// MI455X (gfx1250) — compile-verified
//
#include <hip/hip_runtime.h>
#include <math.h>

// Problem constants
#define BS   128
#define NENT 512
#define N1   513      // N+1
#define DENT 256
#define KK   128      // K
#define DIN  1024
#define DF   256
#define HH   128
#define LL   2
#define SS   64
#define NEGV (-1e9f)

typedef __attribute__((ext_vector_type(2))) float v2f;
typedef __attribute__((ext_vector_type(8))) float v8f;

// ---------------------------------------------------------------------------
// Kernel A: key[b,n,k] = entity_emb[b,n,:] . key_w[k,:] + key_b[k]   (n < 512)
// fp32 WMMA 16x16x4. Block = 256 thr (8 waves): 32 rows x 128 cols.
// Each wave: 2 row-tiles x 1 col-tile -> B fragment reused by 2 WMMAs.
// ---------------------------------------------------------------------------
__global__ __launch_bounds__(256) void k_key_gemm(
    const float* __restrict__ ee, const float* __restrict__ kw,
    const float* __restrict__ kb, float* __restrict__ key)
{
    __shared__ float lA[32 * 260];                 // 32 x 256, stride 260 (bank spread)
    const int t = threadIdx.x;
    const int wave = t >> 5, lane = t & 31;
    const int row0 = blockIdx.x * 32;              // global row in [0, BS*NENT)

    { // stage A tile: 32 x 256 floats, 32 per thread
        const int r = t >> 3, c0 = (t & 7) * 32;
        const float* src = ee + (size_t)(row0 + r) * DENT + c0;
        float* dst = lA + r * 260 + c0;
        #pragma unroll
        for (int j = 0; j < 32; ++j) dst[j] = src[j];
    }
    __syncthreads();

    const int col0 = wave * 16;
    const int mn   = lane & 15;                    // A-row (m) == B-col (n) lane index
    const int klo  = (lane >> 4) * 2;
    const float* brow = kw + (size_t)(col0 + mn) * DENT;   // B[d][col] = key_w[col][d]
    const float* a0p = lA + mn * 260 + klo;
    const float* a1p = lA + (16 + mn) * 260 + klo;
    v8f acc0 = {}, acc1 = {};
    #pragma unroll 8
    for (int k0 = 0; k0 < DENT; k0 += 4) {
        v2f a0, a1, b;
        b.x  = brow[k0 + klo]; b.y  = brow[k0 + klo + 1];
        a0.x = a0p[k0];        a0.y = a0p[k0 + 1];
        a1.x = a1p[k0];        a1.y = a1p[k0 + 1];
        acc0 = __builtin_amdgcn_wmma_f32_16x16x4_f32(false, a0, false, b,
                                                     (short)0, acc0, false, false);
        acc1 = __builtin_amdgcn_wmma_f32_16x16x4_f32(false, a1, false, b,
                                                     (short)0, acc1, false, false);
    }
    const float bias = kb[col0 + mn];
    const int mbase = (lane >> 4) * 8;
    #pragma unroll
    for (int v = 0; v < 8; ++v) {
        int gr = row0 + mbase + v;
        int bi = gr >> 9, ni = gr & 511;
        key[((size_t)bi * N1 + ni) * KK + col0 + mn] = acc0[v] + bias;
        gr += 16; bi = gr >> 9; ni = gr & 511;
        key[((size_t)bi * N1 + ni) * KK + col0 + mn] = acc1[v] + bias;
    }
}

// Row N=512 -> 0 ; row entity_num[b] -> end_embedding
__global__ void k_key_fix(const int* __restrict__ en, const float* __restrict__ endv,
                          float* __restrict__ key)
{
    const int b = blockIdx.x, t = threadIdx.x;
    key[((size_t)b * N1 + NENT) * KK + t] = 0.f;
    key[((size_t)b * N1 + en[b]) * KK + t] = endv[t];
}

// ---------------------------------------------------------------------------
// Kernel B1: H1[g,:] = relu( (key[b,su]/512) @ e1_w^T + e1_b ), g = b*S+i
// M=8192, cols=256, inner=128. Block: 32 rows x 128 cols, grid (256,2).
// ---------------------------------------------------------------------------
__global__ __launch_bounds__(256) void k_emb_mlp1(
    const float* __restrict__ key, const int* __restrict__ su,
    const float* __restrict__ w1, const float* __restrict__ b1,
    float* __restrict__ H1)
{
    __shared__ float lA[32 * 132];
    const int t = threadIdx.x;
    const int wave = t >> 5, lane = t & 31;
    const int row0 = blockIdx.x * 32;

    { // gather + scale A tile: 32 x 128, 16 per thread
        const int r = t >> 3, c0 = (t & 7) * 16;
        const int g = row0 + r, bb = g >> 6, ii = g & 63;
        const float* src = key + ((size_t)bb * N1 + su[bb * SS + ii]) * KK + c0;
        float* dst = lA + r * 132 + c0;
        #pragma unroll
        for (int j = 0; j < 16; ++j) dst[j] = src[j] * (1.0f / 512.0f);
    }
    __syncthreads();

    const int col0 = (blockIdx.y * 8 + wave) * 16;
    const int mn = lane & 15;
    const int klo = (lane >> 4) * 2;
    const float* brow = w1 + (size_t)(col0 + mn) * KK;
    const float* a0p = lA + mn * 132 + klo;
    const float* a1p = lA + (16 + mn) * 132 + klo;
    v8f acc0 = {}, acc1 = {};
    #pragma unroll 8
    for (int k0 = 0; k0 < KK; k0 += 4) {
        v2f a0, a1, b;
        b.x  = brow[k0 + klo]; b.y  = brow[k0 + klo + 1];
        a0.x = a0p[k0];        a0.y = a0p[k0 + 1];
        a1.x = a1p[k0];        a1.y = a1p[k0 + 1];
        acc0 = __builtin_amdgcn_wmma_f32_16x16x4_f32(false, a0, false, b,
                                                     (short)0, acc0, false, false);
        acc1 = __builtin_amdgcn_wmma_f32_16x16x4_f32(false, a1, false, b,
                                                     (short)0, acc1, false, false);
    }
    const float bias = b1[col0 + mn];
    const int mbase = (lane >> 4) * 8;
    #pragma unroll
    for (int v = 0; v < 8; ++v) {
        const int g0 = row0 + mbase + v;
        float x0 = acc0[v] + bias;
        H1[(size_t)g0 * DF + col0 + mn] = x0 > 0.f ? x0 : 0.f;
        float x1 = acc1[v] + bias;
        H1[(size_t)(g0 + 16) * DF + col0 + mn] = x1 > 0.f ? x1 : 0.f;
    }
}

// ---------------------------------------------------------------------------
// Kernel B2: key_emb_sel[g,:] = H1[g,:] @ e2_w^T + e2_b
// M=8192, cols=1024, inner=256. Block: 32 rows x 128 cols, grid (256,8).
// ---------------------------------------------------------------------------
__global__ __launch_bounds__(256) void k_emb_mlp2(
    const float* __restrict__ H1, const float* __restrict__ w2,
    const float* __restrict__ b2, float* __restrict__ kes)
{
    __shared__ float lA[32 * 260];
    const int t = threadIdx.x;
    const int wave = t >> 5, lane = t & 31;
    const int row0 = blockIdx.x * 32;

    {
        const int r = t >> 3, c0 = (t & 7) * 32;
        const float* src = H1 + (size_t)(row0 + r) * DF + c0;
        float* dst = lA + r * 260 + c0;
        #pragma unroll
        for (int j = 0; j < 32; ++j) dst[j] = src[j];
    }
    __syncthreads();

    const int col0 = (blockIdx.y * 8 + wave) * 16;
    const int mn = lane & 15;
    const int klo = (lane >> 4) * 2;
    const float* brow = w2 + (size_t)(col0 + mn) * DF;
    const float* a0p = lA + mn * 260 + klo;
    const float* a1p = lA + (16 + mn) * 260 + klo;
    v8f acc0 = {}, acc1 = {};
    #pragma unroll 8
    for (int k0 = 0; k0 < DF; k0 += 4) {
        v2f a0, a1, b;
        b.x  = brow[k0 + klo]; b.y  = brow[k0 + klo + 1];
        a0.x = a0p[k0];        a0.y = a0p[k0 + 1];
        a1.x = a1p[k0];        a1.y = a1p[k0 + 1];
        acc0 = __builtin_amdgcn_wmma_f32_16x16x4_f32(false, a0, false, b,
                                                     (short)0, acc0, false, false);
        acc1 = __builtin_amdgcn_wmma_f32_16x16x4_f32(false, a1, false, b,
                                                     (short)0, acc1, false, false);
    }
    const float bias = b2[col0 + mn];
    const int mbase = (lane >> 4) * 8;
    #pragma unroll
    for (int v = 0; v < 8; ++v) {
        const int g0 = row0 + mbase + v;
        kes[(size_t)g0 * DIN + col0 + mn] = acc0[v] + bias;
        kes[(size_t)(g0 + 16) * DIN + col0 + mn] = acc1[v] + bias;
    }
}

// ---------------------------------------------------------------------------
// Kernel C: persistent per-batch scan (one workgroup of 256 per batch elem).
// ---------------------------------------------------------------------------
__device__ __forceinline__ float sigmf(float x) { return 1.f / (1.f + __expf(-x)); }

__device__ __forceinline__ float block_sum_256(float v, float* red, int t) {
    red[t] = v; __syncthreads();
    #pragma unroll
    for (int off = 128; off > 0; off >>= 1) {
        if (t < off) red[t] += red[t + off];
        __syncthreads();
    }
    float s = red[0];
    __syncthreads();
    return s;
}

__global__ __launch_bounds__(256) void k_scan(
    const float* __restrict__ emb,
    const float* __restrict__ q1w, const float* __restrict__ q1b,
    const float* __restrict__ q2w, const float* __restrict__ q2b,
    const float* __restrict__ wih, const float* __restrict__ whh,
    const float* __restrict__ lnig, const float* __restrict__ lnib,
    const float* __restrict__ lnhg, const float* __restrict__ lnhb,
    const float* __restrict__ lncg, const float* __restrict__ lncb,
    const int* __restrict__ entity_num, const int* __restrict__ sel_units,
    const int* __restrict__ sel_num,
    const float* __restrict__ key, const float* __restrict__ kes,
    float* __restrict__ out_logits, float* __restrict__ out_ae,
    float* __restrict__ out_sn)
{
    __shared__ float ae[DIN];
    __shared__ float x1[DF];
    __shared__ float xc[HH];
    __shared__ float s_gi[512];
    __shared__ float s_gh[512];
    __shared__ float hbuf[LL][HH];
    __shared__ float cbuf[LL][HH];
    __shared__ float red[256];
    __shared__ unsigned char selflag[N1];

    const int b = blockIdx.x, t = threadIdx.x;
    for (int j = t; j < DIN; j += 256) ae[j] = emb[(size_t)b * DIN + j];
    if (t < HH) { hbuf[0][t] = 0.f; hbuf[1][t] = 0.f; cbuf[0][t] = 0.f; cbuf[1][t] = 0.f; }
    for (int j = t; j < N1; j += 256) selflag[j] = 0;
    const int en = entity_num[b];
    const int snum = sel_num[b];
    __syncthreads();

    for (int i = 0; i < SS; ++i) {
        // ---- query MLP: x1 = relu(ae @ q1w^T + q1b)  (256 dots of 1024)
        {
            const float* wr = q1w + (size_t)t * DIN;
            float s = 0.f;
            for (int k2 = 0; k2 < DIN; ++k2) s += ae[k2] * wr[k2];
            s += q1b[t];
            x1[t] = s > 0.f ? s : 0.f;
        }
        __syncthreads();
        // ---- x = x1 @ q2w^T + q2b  (128 dots of 256)
        if (t < KK) {
            const float* wr = q2w + (size_t)t * DF;
            float s = 0.f;
            for (int k2 = 0; k2 < DF; ++k2) s += x1[k2] * wr[k2];
            xc[t] = s + q2b[t];
        }
        __syncthreads();

        // ---- LayerNorm-LSTM stack
        for (int l = 0; l < LL; ++l) {
            const float* WI = wih + (size_t)l * 512 * HH;
            const float* WH = whh + (size_t)l * 512 * HH;
            #pragma unroll
            for (int jj = 0; jj < 2; ++jj) {
                const int j = t + jj * 256;
                const float* wi = WI + (size_t)j * HH;
                const float* wh = WH + (size_t)j * HH;
                float si = 0.f, sh = 0.f;
                for (int k2 = 0; k2 < HH; ++k2) {
                    si += xc[k2] * wi[k2];
                    sh += hbuf[l][k2] * wh[k2];
                }
                s_gi[j] = si; s_gh[j] = sh;
            }
            __syncthreads();
            const float a0 = s_gi[t], a1 = s_gi[t + 256];
            const float b0 = s_gh[t], b1 = s_gh[t + 256];
            const float mi = block_sum_256(a0 + a1, red, t) * (1.f / 512.f);
            const float vi = block_sum_256((a0 - mi) * (a0 - mi) + (a1 - mi) * (a1 - mi),
                                           red, t) * (1.f / 512.f);
            const float ri = rsqrtf(vi + 1e-5f);
            const float mh = block_sum_256(b0 + b1, red, t) * (1.f / 512.f);
            const float vh = block_sum_256((b0 - mh) * (b0 - mh) + (b1 - mh) * (b1 - mh),
                                           red, t) * (1.f / 512.f);
            const float rh = rsqrtf(vh + 1e-5f);
            const int o0 = l * 512 + t, o1 = o0 + 256;
            s_gi[t]       = (a0 - mi) * ri * lnig[o0] + lnib[o0]
                          + (b0 - mh) * rh * lnhg[o0] + lnhb[o0];
            s_gi[t + 256] = (a1 - mi) * ri * lnig[o1] + lnib[o1]
                          + (b1 - mh) * rh * lnhg[o1] + lnhb[o1];
            __syncthreads();
            float cn = 0.f;
            if (t < HH) {
                const float igv = sigmf(s_gi[t]);
                const float fgv = sigmf(s_gi[HH + t]);
                const float ggv = tanhf(s_gi[2 * HH + t]);
                cn = fgv * cbuf[l][t] + igv * ggv;
            }
            const float mc = block_sum_256((t < HH) ? cn : 0.f, red, t) * (1.f / 128.f);
            const float vc = block_sum_256((t < HH) ? (cn - mc) * (cn - mc) : 0.f,
                                           red, t) * (1.f / 128.f);
            const float rc = rsqrtf(vc + 1e-5f);
            if (t < HH) {
                const float cf = (cn - mc) * rc * lncg[l * HH + t] + lncb[l * HH + t];
                cbuf[l][t] = cf;
                const float hv = sigmf(s_gi[3 * HH + t]) * tanhf(cf);
                hbuf[l][t] = hv;
                xc[t] = hv;
            }
            __syncthreads();
        }

        // ---- logits: masked dot(h, key[b,n,:])
        const float* kbase = key + (size_t)b * N1 * KK;
        for (int n = t; n < N1; n += 256) {
            const bool m = (n <= en) && (selflag[n] == 0) && !((i == 0) && (n == en));
            float s = NEGV;
            if (m) {
                const float* kr = kbase + (size_t)n * KK;
                float d = 0.f;
                for (int k2 = 0; k2 < KK; ++k2) d += xc[k2] * kr[k2];
                s = d;
            }
            out_logits[((size_t)b * SS + i) * N1 + n] = s;
        }
        __syncthreads();

        // ---- ae accumulation + selected-flag update
        const int su_i = sel_units[b * SS + i];
        const bool gate = ((i + 1) <= snum) && (su_i != en);
        if (gate) {
            const float* ke = kes + ((size_t)b * SS + i) * DIN;
            for (int j = t; j < DIN; j += 256) ae[j] += ke[j];
        }
        if (t == 0) selflag[su_i] = 1;
        __syncthreads();
    }

    for (int j = t; j < DIN; j += 256) out_ae[(size_t)b * DIN + j] = ae[j];
    if (t == 0) out_sn[b] = (float)snum;
}

// ---------------------------------------------------------------------------
extern "C" void kernel_launch(void* const* d_in, const int* in_sizes, int n_in,
                              void* d_out, int out_size, void* d_ws, size_t ws_size,
                              hipStream_t stream)
{
    const float* embedding  = (const float*)d_in[0];
    const float* entity_emb = (const float*)d_in[1];
    const float* key_w      = (const float*)d_in[2];
    const float* key_b      = (const float*)d_in[3];
    const float* q1_w       = (const float*)d_in[4];
    const float* q1_b       = (const float*)d_in[5];
    const float* q2_w       = (const float*)d_in[6];
    const float* q2_b       = (const float*)d_in[7];
    const float* e1_w       = (const float*)d_in[8];
    const float* e1_b       = (const float*)d_in[9];
    const float* e2_w       = (const float*)d_in[10];
    const float* e2_b       = (const float*)d_in[11];
    const float* end_emb    = (const float*)d_in[12];
    const float* lstm_wih   = (const float*)d_in[13];
    const float* lstm_whh   = (const float*)d_in[14];
    const float* ln_ig      = (const float*)d_in[15];
    const float* ln_ib      = (const float*)d_in[16];
    const float* ln_hg      = (const float*)d_in[17];
    const float* ln_hb      = (const float*)d_in[18];
    const float* ln_cg      = (const float*)d_in[19];
    const float* ln_cb      = (const float*)d_in[20];
    const int*   entity_num = (const int*)d_in[21];
    const int*   sel_units  = (const int*)d_in[22];
    const int*   sel_num    = (const int*)d_in[23];

    float* ws  = (float*)d_ws;
    float* key = ws;                                  // 128*513*128   = 8,404,992
    float* H1  = ws + 8404992;                        // 8192*256      = 2,097,152
    float* kes = ws + 8404992 + 2097152;              // 128*64*1024   = 8,388,608

    float* out = (float*)d_out;
    float* out_logits = out;                          // 128*64*513 = 4,202,496
    float* out_ae     = out + 4202496;                // 128*1024   =   131,072
    float* out_sn     = out + 4202496 + 131072;       // 128

    k_key_gemm<<<dim3((BS * NENT) / 32), 256, 0, stream>>>(entity_emb, key_w, key_b, key);
    k_key_fix<<<dim3(BS), 128, 0, stream>>>(entity_num, end_emb, key);
    k_emb_mlp1<<<dim3((BS * SS) / 32, 2), 256, 0, stream>>>(key, sel_units, e1_w, e1_b, H1);
    k_emb_mlp2<<<dim3((BS * SS) / 32, 8), 256, 0, stream>>>(H1, e2_w, e2_b, kes);
    k_scan<<<dim3(BS), 256, 0, stream>>>(embedding, q1_w, q1_b, q2_w, q2_b,
                                         lstm_wih, lstm_whh,
                                         ln_ig, ln_ib, ln_hg, ln_hb, ln_cg, ln_cb,
                                         entity_num, sel_units, sel_num,
                                         key, kes, out_logits, out_ae, out_sn);
}